// DotProductAttention_70746701300156
// MI455X (gfx1250) — compile-verified
//
#include <hip/hip_runtime.h>
#include <math.h>

#define B_ 32
#define S_ 4096
#define H_ 1024
#define ROWS 32
#define CHUNKS (S_ / ROWS)      // 128 chunks per batch
// TDM pad: +2 dwords after every 256 stored dwords -> row stride 1032 dwords,
// element (r,c) at dword  r*1032 + c + 2*(c>>8).  Keeps b64 operands 8B-aligned
// and gives an 8-dword per-row bank rotation (<=2-way conflicts on WMMA B loads).
#define LDK 1032

typedef __attribute__((ext_vector_type(2))) float v2f;
typedef __attribute__((ext_vector_type(8))) float v8f;
typedef __attribute__((ext_vector_type(4))) unsigned int v4u;
typedef __attribute__((ext_vector_type(4))) int v4i;
typedef __attribute__((ext_vector_type(8))) int v8i;

static __device__ inline unsigned int lds_byte_addr(const void* p) {
    // generic -> LDS address space; AS(3) pointers are 32-bit LDS byte offsets
    return (unsigned int)(uintptr_t)(__attribute__((address_space(3))) const void*)p;
}

// ---------------------------------------------------------------------------
// Kernel 1: per (batch, 32-row chunk):
//   - TDM tensor_load_to_lds DMAs the 32x1024 f32 keys tile into LDS
//     (hardware pad inserts the bank-rotation padding), single HBM read
//   - scores via v_wmma_f32_16x16x4_f32 (8 waves = 2 row-groups x 4 K-quarters)
//   - chunk-local softmax stats (m_i, l_i) + raw scores to workspace
//   - partial context from the same LDS tile
// ---------------------------------------------------------------------------
__global__ __launch_bounds__(256) void attn_fused_chunk(
    const float* __restrict__ q,        // (B, 1, H)
    const float* __restrict__ keys,     // (B, S, H)
    float* __restrict__ scoresWS,       // (B, S) raw scores
    float* __restrict__ mWS,            // (B, CHUNKS)
    float* __restrict__ lWS,            // (B, CHUNKS)
    float* __restrict__ partWS)         // (B, CHUNKS, H)
{
    __shared__ float tileK[ROWS * LDK];  // 32*1032*4 = 132096 B  (2 blocks/WGP)
    __shared__ float qs[H_];
    __shared__ float sc[ROWS];
    __shared__ float red[256];

    const int tid   = threadIdx.x;
    const int b     = blockIdx.y;
    const int chunk = blockIdx.x;
    const int row0  = chunk * ROWS;

    // ---- issue TDM: keys tile -> LDS (EXEC-independent DMA, TENSORcnt) ----
    if (tid == 0) {
        const unsigned long long gaddr =
            (unsigned long long)(uintptr_t)(keys + ((size_t)b * S_ + row0) * H_);
        const unsigned int lds = lds_byte_addr(&tileK[0]);

        v4u g0;
        g0[0] = 1u;                                        // count=1 (valid), user mode
        g0[1] = lds;                                       // lds_addr (bytes)
        g0[2] = (unsigned int)gaddr;                       // global_addr[31:0]
        g0[3] = (unsigned int)((gaddr >> 32) & 0x01FFFFFFu) | (2u << 30); // [56:32] | type=2

        unsigned int w[8];
        w[0] = (2u << 16)            // data_size = 4 bytes
             | (1u << 20)            // pad_enable
             | (7u << 22)            // pad_interval: 256 dwords
             | (1u << 25);           // pad_amount:   2 dwords
        w[1] = (unsigned int)H_ << 16;      // tensor_dim0[15:0]=1024 in [31:16]
        w[2] = (unsigned int)ROWS << 16;    // tensor_dim0[31:16]=0 | tensor_dim1[15:0]=32
        w[3] = (unsigned int)H_ << 16;      // tensor_dim1[31:16]=0 | tile_dim0=1024
        w[4] = (unsigned int)ROWS;          // tile_dim1=32 | tile_dim2=0
        w[5] = (unsigned int)H_;            // tensor_dim0_stride[31:0]=1024
        w[6] = 0u;                          // dim0_stride[47:32]=0 | dim1_stride[15:0]=0
        w[7] = 0u;                          // dim1_stride[47:16]=0
        v8i g1;
        #pragma unroll
        for (int i = 0; i < 8; ++i) g1[i] = (int)w[i];

        v4i g2 = {0, 0, 0, 0};              // 2D tile: groups 2/3 unused
        v4i g3 = {0, 0, 0, 0};
        v8i g4 = {0, 0, 0, 0, 0, 0, 0, 0};  // clang-23 6-arg form: extra group, zero-filled

        __builtin_amdgcn_tensor_load_to_lds(g0, g1, g2, g3, g4, 0);
    }

    // ---- query -> LDS with plain vector loads (tiny), init sc ----
    {
        const float4* q4 = (const float4*)(q + (size_t)b * H_);
        float4 qq = q4[tid];                      // 256 threads * 4 = 1024 floats
        *(float4*)&qs[tid * 4] = qq;
        if (tid < ROWS) sc[tid] = 0.0f;
    }
    if (tid == 0) __builtin_amdgcn_s_wait_tensorcnt(0);   // DMA complete
    __syncthreads();

    // ---- WMMA score phase: wave w -> row-group (w&1), K-quarter (w>>1) ----
    // D(16x16) = A(16x4: all rows = q[k..k+3]) * B(4x16: 16 key rows) + C
    // 32-bit operand layout: lanes 0-15 hold K={0(V0),1(V1)}, lanes 16-31 K={2,3}.
    {
        const int w    = tid >> 5;
        const int lane = tid & 31;
        const int half = lane >> 4;               // 0: K=0,1   1: K=2,3
        const int n    = lane & 15;               // B column = key row within group
        const int g    = w & 1;                   // row group (rows g*16 .. g*16+15)
        const int kq   = w >> 1;                  // K quarter (256 wide)
        const int k0   = kq << 8;
        const int koff = half * 2;
        const float* arow = &qs[koff];
        const float* brow = &tileK[(g * 16 + n) * LDK + koff + 2 * kq]; // +pad adjust

        v8f c = {};
        #pragma unroll 8
        for (int k = k0; k < k0 + 256; k += 4) {
            v2f a  = *(const v2f*)(arow + k);
            v2f bm = *(const v2f*)(brow + k);
            c = __builtin_amdgcn_wmma_f32_16x16x4_f32(
                    false, a, false, bm, (short)0, c, false, false);
        }
        // D rows identical; lane L<16: c[0] = partial score of key row (g*16+L)
        if (lane < 16) atomicAdd(&sc[g * 16 + lane], c[0]);  // combine 4 K-quarters
    }
    __syncthreads();

    // ---- raw scores out + chunk-local softmax stats ----
    if (tid < ROWS) scoresWS[(size_t)b * S_ + row0 + tid] = sc[tid];

    float v = (tid < ROWS) ? sc[tid] : -INFINITY;
    red[tid] = v; __syncthreads();
    for (int s = 128; s > 0; s >>= 1) {
        if (tid < s) red[tid] = fmaxf(red[tid], red[tid + s]);
        __syncthreads();
    }
    const float m = red[0];
    __syncthreads();

    const float e = (tid < ROWS) ? __expf(sc[tid] - m) : 0.0f;
    red[tid] = e; __syncthreads();
    for (int s = 128; s > 0; s >>= 1) {
        if (tid < s) red[tid] += red[tid + s];
        __syncthreads();
    }
    if (tid == 0) {
        mWS[b * CHUNKS + chunk] = m;
        lWS[b * CHUNKS + chunk] = red[0];
    }
    __syncthreads();
    if (tid < ROWS) sc[tid] = e;                  // local (unnormalized) weights
    __syncthreads();

    // ---- partial context from the LDS tile (no second HBM read) ----
    {
        const int h0 = tid * 4;                   // 256 threads x 4 cols = 1024
        const float* col = &tileK[h0 + 2 * (h0 >> 8)];   // pad-adjusted column base
        float ax = 0.f, ay = 0.f, az = 0.f, aw = 0.f;
        #pragma unroll 8
        for (int s = 0; s < ROWS; ++s) {
            const float wv = sc[s];               // LDS broadcast
            const v2f k01 = *(const v2f*)(col + s * LDK);      // 8B-aligned
            const v2f k23 = *(const v2f*)(col + s * LDK + 2);
            ax += wv * k01[0]; ay += wv * k01[1];
            az += wv * k23[0]; aw += wv * k23[1];
        }
        float4 r; r.x = ax; r.y = ay; r.z = az; r.w = aw;
        *(float4*)(partWS + ((size_t)(b * CHUNKS + chunk)) * H_ + h0) = r;
    }
}

// ---------------------------------------------------------------------------
// Kernel 2: per batch — merge online-softmax stats, emit weights + context.
// d_out layout: [0, B*H) context, [B*H, B*H + B*S) attention weights.
// ---------------------------------------------------------------------------
__global__ __launch_bounds__(256) void attn_finalize(
    const float* __restrict__ scoresWS,
    const float* __restrict__ mWS,
    const float* __restrict__ lWS,
    const float* __restrict__ partWS,
    float* __restrict__ out)
{
    __shared__ float red[256];
    __shared__ float fac[CHUNKS];

    const int tid = threadIdx.x;
    const int b   = blockIdx.x;

    const float mv = (tid < CHUNKS) ? mWS[b * CHUNKS + tid] : -INFINITY;
    red[tid] = mv; __syncthreads();
    for (int s = 128; s > 0; s >>= 1) {
        if (tid < s) red[tid] = fmaxf(red[tid], red[tid + s]);
        __syncthreads();
    }
    const float m = red[0];
    __syncthreads();

    const float fv = (tid < CHUNKS) ? __expf(mv - m) : 0.0f;
    red[tid] = (tid < CHUNKS) ? fv * lWS[b * CHUNKS + tid] : 0.0f;
    __syncthreads();
    for (int s = 128; s > 0; s >>= 1) {
        if (tid < s) red[tid] += red[tid + s];
        __syncthreads();
    }
    const float L = red[0];
    __syncthreads();
    if (tid < CHUNKS) fac[tid] = fv;
    __syncthreads();

    const float invL = 1.0f / L;

    // attention weights: exp(score - m) / L
    float* wout = out + (size_t)B_ * H_;
    for (int s = tid; s < S_; s += 256)
        wout[(size_t)b * S_ + s] = __expf(scoresWS[(size_t)b * S_ + s] - m) * invL;

    // context: (sum_i fac[i] * partial_i) / L
    const int h0 = tid * 4;
    float ax = 0.f, ay = 0.f, az = 0.f, aw = 0.f;
    #pragma unroll 8
    for (int i = 0; i < CHUNKS; ++i) {
        const float fc = fac[i];
        const float4 p = *(const float4*)(partWS + ((size_t)(b * CHUNKS + i)) * H_ + h0);
        ax += fc * p.x; ay += fc * p.y; az += fc * p.z; aw += fc * p.w;
    }
    float4 r; r.x = ax * invL; r.y = ay * invL; r.z = az * invL; r.w = aw * invL;
    *(float4*)(out + (size_t)b * H_ + h0) = r;
}

extern "C" void kernel_launch(void* const* d_in, const int* in_sizes, int n_in,
                              void* d_out, int out_size, void* d_ws, size_t ws_size,
                              hipStream_t stream) {
    const float* q    = (const float*)d_in[0];   // (32, 1, 1024)
    const float* keys = (const float*)d_in[1];   // (32, 4096, 1024)
    float* out = (float*)d_out;

    float* ws       = (float*)d_ws;
    float* scoresWS = ws;                                  // B*S        = 131072 f
    float* mWS      = scoresWS + (size_t)B_ * S_;          // B*CHUNKS   =   4096 f
    float* lWS      = mWS + (size_t)B_ * CHUNKS;           // B*CHUNKS   =   4096 f
    float* partWS   = lWS + (size_t)B_ * CHUNKS;           // B*CHUNKS*H = 4194304 f (~17.3MB total)

    dim3 g1(CHUNKS, B_);                                   // 128 x 32 = 4096 blocks
    attn_fused_chunk<<<g1, 256, 0, stream>>>(q, keys, scoresWS, mWS, lWS, partWS);
    attn_finalize<<<B_, 256, 0, stream>>>(scoresWS, mWS, lWS, partWS, out);
}